// monet_polar_regression_confounded_batched_31782757990676
// MI455X (gfx1250) — compile-verified
//
#include <hip/hip_runtime.h>
#include <hip/hip_bf16.h>

typedef float v2f __attribute__((ext_vector_type(2)));
typedef float v8f __attribute__((ext_vector_type(8)));

__device__ __forceinline__ int imin(int a, int b) { return a < b ? a : b; }

// ---------------------------------------------------------------------------
// fp32 GEMM tile kernel: one wave32 computes one 16x16 tile of
// C = A(MxK) * B(KxN) [+ bias] [relu], via V_WMMA_F32_16X16X4_F32.
// Fragment layouts per CDNA5 ISA 7.12.2:
//   A 16x4 : lane L holds row m=L&15; K pair = (L>>4)*2 in 2 VGPRs
//   B 4x16 : lane L holds col n=L&15; K pair = (L>>4)*2 in 2 VGPRs
//   C 16x16: lane L holds col n=L&15; VGPR r holds row m = r + (L>>4)*8
// Out-of-range rows/cols are CLAMPED (not masked): their products land only
// in C entries the store loop suppresses, so the hot loop has zero branches
// and all loads are unconditional (no exec-mask save/restore).
// ---------------------------------------------------------------------------
__global__ __launch_bounds__(32)
void monet_wmma_gemm_f32(const float* __restrict__ A, const float* __restrict__ Bm,
                         const float* __restrict__ bias, float* __restrict__ C,
                         int M, int N, int Kdim, int tilesN, int relu)
{
    const int tile = blockIdx.x;
    const int tm = tile / tilesN;
    const int tn = tile % tilesN;
    const int lane = threadIdx.x & 31;
    const int half = lane >> 4;       // 0/1
    const int l16  = lane & 15;

    const int row = tm * 16 + l16;    // logical A row / C row-base
    const int col = tn * 16 + l16;    // logical B col / C col
    const int am = imin(row, M - 1);  // clamped (garbage rows never stored)
    const int bn = imin(col, N - 1);  // clamped (garbage cols never stored)

    const float* Arow = A + (size_t)am * Kdim;
    const float* Bcol = Bm + bn;

    const int kfull = Kdim & ~3;
    v8f acc = {};
    for (int k0 = 0; k0 < kfull; k0 += 4) {
        const int ka = k0 + half * 2;
        v2f a = *reinterpret_cast<const v2f*>(Arow + ka);  // 8B-aligned pair
        v2f b;
        b.x = Bcol[(size_t)(ka + 0) * N];
        b.y = Bcol[(size_t)(ka + 1) * N];
        acc = __builtin_amdgcn_wmma_f32_16x16x4_f32(
            false, a, false, b, (short)0, acc, false, false);
    }
    if (kfull < Kdim) {  // uniform branch; only the cin==2 level takes it
        const int ka  = kfull + half * 2;
        const int k0c = imin(ka + 0, Kdim - 1);
        const int k1c = imin(ka + 1, Kdim - 1);
        const float a0 = Arow[k0c];
        const float a1 = Arow[k1c];
        const float b0 = Bcol[(size_t)k0c * N];
        const float b1 = Bcol[(size_t)k1c * N];
        v2f a, b;
        a.x = (ka + 0 < Kdim) ? a0 : 0.0f;  // per-lane select, no branch
        a.y = (ka + 1 < Kdim) ? a1 : 0.0f;
        b.x = (ka + 0 < Kdim) ? b0 : 0.0f;
        b.y = (ka + 1 < Kdim) ? b1 : 0.0f;
        acc = __builtin_amdgcn_wmma_f32_16x16x4_f32(
            false, a, false, b, (short)0, acc, false, false);
    }

    float bv = 0.0f;
    if (bias) bv = bias[bn];
#pragma unroll
    for (int r = 0; r < 8; ++r) {
        const int cm = tm * 16 + r + half * 8;
        if (cm < M && col < N) {
            float v = acc[r] + bv;
            if (relu) v = fmaxf(v, 0.0f);
            C[(size_t)cm * N + col] = v;
        }
    }
}

// ---------------------------------------------------------------------------
// Per-edge Gaussian mixture weights: w[e,k] = exp(-0.5 * sum_d (p-mu)^2/(s^2+eps))
// ---------------------------------------------------------------------------
__global__ void monet_edge_w(const float* __restrict__ pseudo,
                             const float* __restrict__ mu,
                             const float* __restrict__ sigma,
                             float* __restrict__ w, int E)
{
    int e = blockIdx.x * blockDim.x + threadIdx.x;
    if (e >= E) return;
    const float p0 = pseudo[e * 2 + 0];
    const float p1 = pseudo[e * 2 + 1];
#pragma unroll
    for (int k = 0; k < 3; ++k) {
        const float d0 = p0 - mu[k * 2 + 0];
        const float d1 = p1 - mu[k * 2 + 1];
        const float s0 = sigma[k * 2 + 0];
        const float s1 = sigma[k * 2 + 1];
        const float q = d0 * d0 / (s0 * s0 + 1e-15f) + d1 * d1 / (s1 * s1 + 1e-15f);
        w[e * 3 + k] = __expf(-0.5f * q);
    }
}

__global__ void monet_zero_f32(float* __restrict__ p, long long n)
{
    long long i = (long long)blockIdx.x * blockDim.x + threadIdx.x;
    if (i < n) p[i] = 0.0f;
}

// ---------------------------------------------------------------------------
// Scatter: agg[dst,c] += sum_k w[e,k] * xg[src, k*cout + c]; cnt[dst] += 1.
// cout is a power of two -> shift/mask instead of div/mod.
// ---------------------------------------------------------------------------
__global__ void monet_edge_scatter(const int* __restrict__ ei,
                                   const float* __restrict__ w,
                                   const float* __restrict__ xg,
                                   float* __restrict__ agg,
                                   float* __restrict__ cnt,
                                   int E, int coutShift)
{
    const int cout = 1 << coutShift;
    long long idx = (long long)blockIdx.x * blockDim.x + threadIdx.x;
    const long long total = (long long)E << coutShift;
    if (idx >= total) return;
    const int e = (int)(idx >> coutShift);
    const int c = (int)(idx & (cout - 1));
    const int src = ei[e];
    const int dst = ei[E + e];
    const float* xgs = xg + (size_t)src * (3 << coutShift);
    __builtin_prefetch(xgs + c, 0, 0);          // global_prefetch_b8
    const float w0 = w[e * 3 + 0];
    const float w1 = w[e * 3 + 1];
    const float w2 = w[e * 3 + 2];
    const float m = w0 * xgs[c] + w1 * xgs[cout + c] + w2 * xgs[2 * cout + c];
    atomicAdd(&agg[((size_t)dst << coutShift) + c], m);
    if (c == 0) atomicAdd(&cnt[dst], 1.0f);
}

// out = relu(agg / max(cnt,1) + xr + bias)
__global__ void monet_node_finalize(const float* __restrict__ agg,
                                    const float* __restrict__ cnt,
                                    const float* __restrict__ xr,
                                    const float* __restrict__ bias,
                                    float* __restrict__ out,
                                    int Nn, int coutShift)
{
    const int cout = 1 << coutShift;
    long long idx = (long long)blockIdx.x * blockDim.x + threadIdx.x;
    const long long total = (long long)Nn << coutShift;
    if (idx >= total) return;
    const int n = (int)(idx >> coutShift);
    const int c = (int)(idx & (cout - 1));
    const float cc = fmaxf(cnt[n], 1.0f);
    const float v = agg[idx] / cc + xr[idx] + bias[c];
    out[idx] = fmaxf(v, 0.0f);
}

// out[b,j,c] = max_{t<7} x[b, hx[j,t], c] for j < L
__global__ void monet_hex_pool(const float* __restrict__ x,
                               const int* __restrict__ hx,
                               float* __restrict__ out,
                               int Nper, int L, int cShift, int Bb)
{
    const int C = 1 << cShift;
    long long idx = (long long)blockIdx.x * blockDim.x + threadIdx.x;
    const long long total = ((long long)Bb * L) << cShift;
    if (idx >= total) return;
    const int c = (int)(idx & (C - 1));
    const long long jb = idx >> cShift;
    const int j = (int)(jb % L);
    const int b = (int)(jb / L);
    const float* xb = x + (((size_t)b * Nper) << cShift);
    float m = -3.402823466e+38f;
#pragma unroll
    for (int t = 0; t < 7; ++t) {
        const int nb = hx[j * 7 + t];
        m = fmaxf(m, xb[((size_t)nb << cShift) + c]);
    }
    out[idx] = m;
}

// xc[b, :41472] = pooled flat; xc[b, 41472+j] = metadata[b]*cw[j] + cb[j]
__global__ void monet_build_xc(const float* __restrict__ pooled,
                               const float* __restrict__ metadata,
                               const float* __restrict__ cw,
                               const float* __restrict__ cb,
                               float* __restrict__ xc)
{
    const int FCIN = 256 * 162 + 4;   // 41476
    int idx = blockIdx.x * blockDim.x + threadIdx.x;
    if (idx >= 4 * FCIN) return;
    const int b = idx / FCIN;
    const int i = idx % FCIN;
    float v;
    if (i < 41472) v = pooled[b * 41472 + i];
    else           v = metadata[b] * cw[i - 41472] + cb[i - 41472];
    xc[idx] = v;
}

// out[b] = dot(h[b,:256], fc2_w) + fc2_b
__global__ void monet_fc2(const float* __restrict__ h,
                          const float* __restrict__ w2,
                          const float* __restrict__ b2,
                          float* __restrict__ out)
{
    int b = threadIdx.x;
    if (b < 4) {
        float s = b2[0];
        for (int c = 0; c < 256; ++c) s += h[b * 256 + c] * w2[c];
        out[b] = s;
    }
}

// ---------------------------------------------------------------------------
// Host orchestration. Input dict order: 0:x, (1,2):(ei6,p6), (3,4):(ei5,p5),
// (5,6):(ei4,p4), (7,8):(ei3,p3), 9..12:hex6..hex3, 13:metadata,
// 14+5i: g,mu,sigma,root,b per level, 34:convm_w, 35:convm_b,
// 36:fc_w, 37:fc_b, 38:fc2_w, 39:fc2_b
// ---------------------------------------------------------------------------
extern "C" void kernel_launch(void* const* d_in, const int* in_sizes, int n_in,
                              void* d_out, int out_size, void* d_ws, size_t ws_size,
                              hipStream_t stream)
{
    (void)in_sizes; (void)n_in; (void)out_size; (void)ws_size;
    const int Bb = 4;
    const int NSh[4]    = {40962, 10242, 2562, 642};
    const int ESh[4]    = {245760, 61440, 15360, 3840};
    const int CINh[4]   = {2, 32, 64, 128};
    const int COUTh[4]  = {32, 64, 128, 256};
    const int CSHIFT[4] = {5, 6, 7, 8};

    float* ws = (float*)d_ws;
    size_t off = 0;
    float* bufA   = ws + off; off += 1310976;   // pooled activations (max 40968*32)
    float* bufB   = ws + off; off += 5242944;   // conv outputs (max 163848*32)
    float* bufXG  = ws + off; off += 15729408;  // x@g (max 163848*96)
    float* bufXR  = ws + off; off += 5242944;   // x@root
    float* bufAGG = ws + off; off += 5242944;   // segment sums
    float* bufCNT = ws + off; off += 163848;    // edge counts
    float* bufW   = ws + off; off += 2949120;   // edge gaussians (max 983040*3)
    float* bufXC  = ws + off; off += 165904;    // fc input (4*41476)
    float* bufH   = ws + off; off += 1024;      // fc hidden (4*256)

    const float* act = (const float*)d_in[0];

    for (int lvl = 0; lvl < 4; ++lvl) {
        const int Nn    = Bb * NSh[lvl];
        const int E     = Bb * ESh[lvl];
        const int cin   = CINh[lvl];
        const int cout  = COUTh[lvl];
        const int cs    = CSHIFT[lvl];
        const int*   ei     = (const int*)  d_in[1 + 2 * lvl];
        const float* pseudo = (const float*)d_in[2 + 2 * lvl];
        const int*   hx     = (const int*)  d_in[9 + lvl];
        const float* g      = (const float*)d_in[14 + 5 * lvl];
        const float* mu     = (const float*)d_in[15 + 5 * lvl];
        const float* sigma  = (const float*)d_in[16 + 5 * lvl];
        const float* root   = (const float*)d_in[17 + 5 * lvl];
        const float* bias   = (const float*)d_in[18 + 5 * lvl];

        // xg = act @ g   (Nn x cin) @ (cin x 3*cout)
        {
            const int Nc = 3 * cout;
            const int tM = (Nn + 15) / 16, tN = (Nc + 15) / 16;
            monet_wmma_gemm_f32<<<tM * tN, 32, 0, stream>>>(
                act, g, nullptr, bufXG, Nn, Nc, cin, tN, 0);
        }
        // xr = act @ root
        {
            const int tM = (Nn + 15) / 16, tN = (cout + 15) / 16;
            monet_wmma_gemm_f32<<<tM * tN, 32, 0, stream>>>(
                act, root, nullptr, bufXR, Nn, cout, cin, tN, 0);
        }
        // gaussian edge weights
        monet_edge_w<<<(E + 255) / 256, 256, 0, stream>>>(pseudo, mu, sigma, bufW, E);
        // zero accumulators
        const long long zn = (long long)Nn * cout;
        monet_zero_f32<<<(int)((zn + 255) / 256), 256, 0, stream>>>(bufAGG, zn);
        monet_zero_f32<<<(Nn + 255) / 256, 256, 0, stream>>>(bufCNT, Nn);
        // scatter messages
        const long long tot = (long long)E * cout;
        monet_edge_scatter<<<(int)((tot + 255) / 256), 256, 0, stream>>>(
            ei, bufW, bufXG, bufAGG, bufCNT, E, cs);
        // mean + root + bias + relu
        monet_node_finalize<<<(int)((zn + 255) / 256), 256, 0, stream>>>(
            bufAGG, bufCNT, bufXR, bias, bufB, Nn, cs);
        // hex max-pool to next level
        const int L = (NSh[lvl] + 6) / 4;
        const long long pt = (long long)Bb * L * cout;
        monet_hex_pool<<<(int)((pt + 255) / 256), 256, 0, stream>>>(
            bufB, hx, bufA, NSh[lvl], L, cs, Bb);
        act = bufA;
    }

    // concat pooled features with metadata linear
    monet_build_xc<<<(4 * 41476 + 255) / 256, 256, 0, stream>>>(
        bufA, (const float*)d_in[13], (const float*)d_in[34],
        (const float*)d_in[35], bufXC);

    // FC1: (4 x 41476) @ (41476 x 256) + bias, relu  -> WMMA
    {
        const int tN = 16;  // 256/16
        monet_wmma_gemm_f32<<<tN, 32, 0, stream>>>(
            bufXC, (const float*)d_in[36], (const float*)d_in[37],
            bufH, 4, 256, 41476, tN, 1);
    }

    // FC2: (4 x 256) @ (256 x 1) + bias
    monet_fc2<<<1, 32, 0, stream>>>(bufH, (const float*)d_in[38],
                                    (const float*)d_in[39], (float*)d_out);
}